// SimpleGNN_88098369175700
// MI455X (gfx1250) — compile-verified
//
#include <hip/hip_runtime.h>
#include <cstdint>
#include <cstddef>

#define NNODES 100000
#define DIN    128
#define DH     48
#define DOUT   40

typedef __attribute__((ext_vector_type(2))) float v2f;
typedef __attribute__((ext_vector_type(8))) float v8f;

// ---------------------------------------------------------------------------
// GEMM1 + bias + ReLU:  h1 = relu(x @ W1 + b1)   [N,128]x[128,48] -> [N,48]
// One wave32 computes one 16x16 output tile with V_WMMA_F32_16X16X4_F32.
// A fragment (16x4 f32): lanes 0-15 hold M=lane, K={kk,kk+1}; lanes 16-31 hold
// M=lane-16, K={kk+2,kk+3}. B fragment (4x16): mirrored, N=lane&15.
// C/D (16x16 f32, 8 VGPRs): VGPR j -> M=j (lanes 0-15) / M=j+8 (lanes 16-31).
// ---------------------------------------------------------------------------
__global__ __launch_bounds__(256) void gemm1_relu_wmma(
    const float* __restrict__ x, const float* __restrict__ W1,
    const float* __restrict__ b1, float* __restrict__ h1)
{
    const int tiles = (NNODES / 16) * (DH / 16);          // 6250 * 3
    int wave = (int)((blockIdx.x * blockDim.x + threadIdx.x) >> 5);
    int lane = (int)(threadIdx.x & 31);
    if (wave >= tiles) return;

    int rt   = wave / (DH / 16);
    int ct   = wave - rt * (DH / 16);
    int row0 = rt << 4;
    int col0 = ct << 4;

    int l15  = lane & 15;
    int hi   = lane >> 4;        // 0 for lanes 0-15, 1 for 16-31
    int koff = hi << 1;          // 0 or 2

    const float* __restrict__ arow = x  + (size_t)(row0 + l15) * DIN;
    const float* __restrict__ bcol = W1 + (size_t)(col0 + l15);

    v8f c = {};
#pragma unroll 8
    for (int kk = 0; kk < DIN; kk += 4) {
        int k0 = kk + koff;
        v2f a, b;
        a.x = arow[k0];
        a.y = arow[k0 + 1];
        b.x = bcol[(size_t)k0 * DH];
        b.y = bcol[(size_t)(k0 + 1) * DH];
        c = __builtin_amdgcn_wmma_f32_16x16x4_f32(
                /*neg_a=*/false, a, /*neg_b=*/false, b,
                /*c_mod=*/(short)0, c, /*reuse_a=*/false, /*reuse_b=*/false);
    }

    int   nc    = col0 + l15;
    float bias  = b1[nc];
    int   mbase = row0 + (hi << 3);
#pragma unroll
    for (int j = 0; j < 8; ++j) {
        float v = c[j] + bias;
        h1[(size_t)(mbase + j) * DH + nc] = v > 0.f ? v : 0.f;
    }
}

// ---------------------------------------------------------------------------
// GEMM2: h2 = (acc1 * inv_deg) @ W2 + b2   [N,48]x[48,40] -> [N,40]
// Mean-aggregation divide fused into the A-fragment load. N padded to 48;
// column tile 2 is half-valid (guarded loads/stores).
// ---------------------------------------------------------------------------
__global__ __launch_bounds__(256) void gemm2_wmma(
    const float* __restrict__ acc1, const float* __restrict__ inv_deg,
    const float* __restrict__ W2, const float* __restrict__ b2,
    float* __restrict__ h2)
{
    const int tiles = (NNODES / 16) * (DH / 16);          // 6250 * 3 (N padded 40->48)
    int wave = (int)((blockIdx.x * blockDim.x + threadIdx.x) >> 5);
    int lane = (int)(threadIdx.x & 31);
    if (wave >= tiles) return;

    int rt   = wave / (DH / 16);
    int ct   = wave - rt * (DH / 16);
    int row0 = rt << 4;
    int col0 = ct << 4;

    int l15  = lane & 15;
    int hi   = lane >> 4;
    int koff = hi << 1;

    int   nc   = col0 + l15;
    bool  nok  = (nc < DOUT);
    float invd = inv_deg[row0 + l15];
    const float* __restrict__ arow = acc1 + (size_t)(row0 + l15) * DH;

    v8f c = {};
#pragma unroll
    for (int kk = 0; kk < DH; kk += 4) {
        int k0 = kk + koff;
        v2f a, b;
        a.x = arow[k0]     * invd;
        a.y = arow[k0 + 1] * invd;
        b.x = nok ? W2[(size_t)k0 * DOUT + nc]       : 0.f;
        b.y = nok ? W2[(size_t)(k0 + 1) * DOUT + nc] : 0.f;
        c = __builtin_amdgcn_wmma_f32_16x16x4_f32(
                false, a, false, b, (short)0, c, false, false);
    }

    if (!nok) return;
    float bias  = b2[nc];
    int   mbase = row0 + (hi << 3);
#pragma unroll
    for (int j = 0; j < 8; ++j) {
        h2[(size_t)(mbase + j) * DOUT + nc] = c[j] + bias;
    }
}

// ---------------------------------------------------------------------------
// Edge scatter kernels (the real bandwidth/atomic cost of this workload).
// ---------------------------------------------------------------------------
__global__ __launch_bounds__(256) void degree_count(
    const int* __restrict__ tgt, float* __restrict__ deg, int nedges)
{
    int e = (int)(blockIdx.x * blockDim.x + threadIdx.x);
    if (e < nedges) atomicAdd(&deg[tgt[e]], 1.0f);
}

__global__ __launch_bounds__(256) void inv_degree(float* __restrict__ deg)
{
    int n = (int)(blockIdx.x * blockDim.x + threadIdx.x);
    if (n < NNODES) deg[n] = 1.0f / fmaxf(deg[n], 1.0f);
}

template<int D>
__global__ __launch_bounds__(256) void scatter_add(
    const int* __restrict__ src, const int* __restrict__ tgt,
    const float* __restrict__ h, float* __restrict__ acc, int nedges)
{
    int g = (int)(blockIdx.x * blockDim.x + threadIdx.x);
    int e = g / D;
    int f = g - e * D;
    if (e >= nedges) return;
    int s = src[e];
    int t = tgt[e];
    atomicAdd(&acc[(size_t)t * D + f], h[(size_t)s * D + f]);
}

// ---------------------------------------------------------------------------
// Fused mean-divide + log_softmax over the 40 classes.
// ---------------------------------------------------------------------------
__global__ __launch_bounds__(256) void mean_log_softmax(
    const float* __restrict__ acc2, const float* __restrict__ inv_deg,
    float* __restrict__ out)
{
    int n = (int)(blockIdx.x * blockDim.x + threadIdx.x);
    if (n >= NNODES) return;
    float invd = inv_deg[n];
    float v[DOUT];
    float mx = -1e30f;
#pragma unroll
    for (int f = 0; f < DOUT; ++f) {
        v[f] = acc2[(size_t)n * DOUT + f] * invd;
        mx = fmaxf(mx, v[f]);
    }
    float s = 0.f;
#pragma unroll
    for (int f = 0; f < DOUT; ++f) s += expf(v[f] - mx);
    float lse = mx + logf(s);
#pragma unroll
    for (int f = 0; f < DOUT; ++f) out[(size_t)n * DOUT + f] = v[f] - lse;
}

// ---------------------------------------------------------------------------
extern "C" void kernel_launch(void* const* d_in, const int* in_sizes, int n_in,
                              void* d_out, int out_size, void* d_ws, size_t ws_size,
                              hipStream_t stream)
{
    const float* x   = (const float*)d_in[0];
    const int*   ei  = (const int*)  d_in[1];
    const float* W1  = (const float*)d_in[2];
    const float* b1  = (const float*)d_in[3];
    const float* W2  = (const float*)d_in[4];
    const float* b2  = (const float*)d_in[5];
    float*       out = (float*)d_out;

    const int nedges = in_sizes[1] / 2;
    const int* src = ei;
    const int* tgt = ei + nedges;

    // Workspace layout: [deg | acc1 | acc2 | h1 | h2]  (zero the first three)
    float* deg  = (float*)d_ws;                       // N
    float* acc1 = deg  + NNODES;                      // N*48
    float* acc2 = acc1 + (size_t)NNODES * DH;         // N*40
    float* h1   = acc2 + (size_t)NNODES * DOUT;       // N*48
    float* h2   = h1   + (size_t)NNODES * DH;         // N*40

    hipMemsetAsync(deg, 0, sizeof(float) * (size_t)NNODES * (1 + DH + DOUT), stream);

    // GEMM1 + ReLU (WMMA)
    {
        int tiles  = (NNODES / 16) * (DH / 16);       // 18750 waves
        int blocks = (tiles * 32 + 255) / 256;
        gemm1_relu_wmma<<<blocks, 256, 0, stream>>>(x, W1, b1, h1);
    }

    // Degrees + inverse
    degree_count<<<(nedges + 255) / 256, 256, 0, stream>>>(tgt, deg, nedges);
    inv_degree<<<(NNODES + 255) / 256, 256, 0, stream>>>(deg);

    // Aggregate h1 into acc1
    {
        long long work = (long long)nedges * DH;
        int blocks = (int)((work + 255) / 256);
        scatter_add<DH><<<blocks, 256, 0, stream>>>(src, tgt, h1, acc1, nedges);
    }

    // GEMM2 with fused mean divide (WMMA)
    {
        int tiles  = (NNODES / 16) * (DH / 16);
        int blocks = (tiles * 32 + 255) / 256;
        gemm2_wmma<<<blocks, 256, 0, stream>>>(acc1, deg, W2, b2, h2);
    }

    // Aggregate h2 into acc2
    {
        long long work = (long long)nedges * DOUT;
        int blocks = (int)((work + 255) / 256);
        scatter_add<DOUT><<<blocks, 256, 0, stream>>>(src, tgt, h2, acc2, nedges);
    }

    // Fused mean + log_softmax
    mean_log_softmax<<<(NNODES + 255) / 256, 256, 0, stream>>>(acc2, deg, out);
}